// MambaBlock_24412594110663
// MI455X (gfx1250) — compile-verified
//
#include <hip/hip_runtime.h>
#include <hip/hip_bf16.h>
#include <math.h>

// ---------------- problem constants (from reference) ----------------
#define D_MODELc 256
#define D_INNERc 512
#define D_STATEc 16
#define D_CONVc  4
#define DT_RANKc 16
#define NBc      8
#define TTc      64
#define JJc      25
#define BSEQc    (NBc*JJc)      // 200 sequences
#define MTOTc    (BSEQc*TTc)    // 12800 tokens
#define E2c      (2*D_INNERc)   // 1024
#define XDBLc    (DT_RANKc + 2*D_STATEc)  // 48

typedef float v2f __attribute__((ext_vector_type(2)));
typedef float v8f __attribute__((ext_vector_type(8)));
typedef int   v4i __attribute__((ext_vector_type(4)));
typedef int   v8i __attribute__((ext_vector_type(8)));

__device__ __forceinline__ float silu_f(float v) { return v / (1.0f + __expf(-v)); }

// token row m -> flat base index into x with (N,T,J,C) layout
__device__ __forceinline__ int x_base(int m) {
    int t  = m & (TTc - 1);
    int bb = m >> 6;          // sequence index b = n*J + j
    int n  = bb / JJc;
    int j  = bb - n * JJc;
    return ((n * TTc + t) * JJc + j) * D_MODELc;
}

__device__ __forceinline__ v8f wmma_f32x4(v2f a, v2f b, v8f c) {
    // V_WMMA_F32_16X16X4_F32: full-f32 matrix op (16x16 tile, K=4)
    return __builtin_amdgcn_wmma_f32_16x16x4_f32(
        /*neg_a=*/false, a, /*neg_b=*/false, b,
        /*c_mod=*/(short)0, c, /*reuse_a=*/false, /*reuse_b=*/false);
}

// Issue a TDM 2D tile load: 512 rows x 16 dwords of W_in K-chunk into LDS.
// D# layout per CDNA5 ISA (group0 128b, group1 256b).
__device__ __forceinline__ void tdm_load_chunk(const float* gsrc, unsigned lds_off) {
    unsigned long long ga = (unsigned long long)(uintptr_t)gsrc;
    v4i g0;
    g0[0] = 1;                                        // count=1 (valid user D#)
    g0[1] = (int)lds_off;                             // lds_addr [63:32]
    g0[2] = (int)(ga & 0xffffffffull);                // global_addr lo
    g0[3] = (int)((ga >> 32) & 0x1ffffffull) | (2 << 30); // global_addr hi | type=2
    v8i g1;
    g1[0] = (2 << 16);        // workgroup_mask=0 | data_size=2 (4B)
    g1[1] = (256 << 16);      // tensor_dim0[15:0]=256 (K extent)
    g1[2] = (1024 << 16);     // tensor_dim0 hi=0 | tensor_dim1[15:0]=1024
    g1[3] = (16 << 16);       // tensor_dim1 hi=0 | tile_dim0=16 dwords
    g1[4] = 512;              // tile_dim1=512 rows | tile_dim2=0
    g1[5] = 256;              // tensor_dim0_stride[31:0]=256 elements
    g1[6] = 0;                // stride0 hi | stride1 lo
    g1[7] = 0;
    asm volatile("tensor_load_to_lds %0, %1"
                 :: "s"(g0), "s"(g1) : "memory");
}

// ---------------- K1: xz = xr @ W_in^T  (M=12800, N=1024, K=256) ----------------
// block = 16 token rows x 512 cols; B K-chunks DMA'd to LDS by the Tensor Data
// Mover (double buffered, overlapped with WMMA); 8 waves x 4 N-tiles, shared A frag.
__global__ void k1_gemm_in(const float* __restrict__ x,
                           const float* __restrict__ W_in,
                           float* __restrict__ xz) {
    extern __shared__ float ldsB[];                  // 2 x (512 rows x 16 dwords) = 64KB
    const int lane  = threadIdx.x & 31;
    const int wave  = threadIdx.x >> 5;
    const int half  = lane >> 4, lm = lane & 15;
    const int nbase = (blockIdx.x & 1) * 512;        // which half of N=1024
    const int m0    = (blockIdx.x >> 1) * 16;

    // A fragment source: rows m0..m0+15 (lane = M), lanes 0-15 K={0,1}, 16-31 K={2,3}
    const float* ap = x + x_base(m0 + lm) + 2 * half;

    v8f c[4];
#pragma unroll
    for (int i = 0; i < 4; ++i) c[i] = (v8f){0.f,0.f,0.f,0.f,0.f,0.f,0.f,0.f};

    // prologue: TDM chunk 0 -> buffer 0
    if (wave == 0)
        tdm_load_chunk(W_in + (size_t)nbase * D_MODELc, 0u);

    const int nloc0 = wave * 64 + lm;                // LDS row of tile 0 for this lane

    for (int kc = 0; kc < 16; ++kc) {                // 16 K-chunks of 16
        if (wave == 0) {
            if (kc + 1 < 16) {
                tdm_load_chunk(W_in + (size_t)nbase * D_MODELc + (kc + 1) * 16,
                               (unsigned)(((kc + 1) & 1) * 32768));
                __builtin_amdgcn_s_wait_tensorcnt(1);  // in-order => chunk kc resident
            } else {
                __builtin_amdgcn_s_wait_tensorcnt(0);
            }
        }
        __syncthreads();                             // buffer kc visible to all waves
        const float* bb = ldsB + (kc & 1) * 8192;    // active buffer
#pragma unroll
        for (int ks = 0; ks < 4; ++ks) {             // 4 WMMA k-steps per chunk
            v2f a = *(const v2f*)(ap + kc * 16 + ks * 4);
#pragma unroll
            for (int i = 0; i < 4; ++i) {
                v2f b = *(const v2f*)(bb + (nloc0 + i * 16) * 16 + ks * 4 + 2 * half);
                c[i] = wmma_f32x4(a, b, c[i]);
            }
        }
        __syncthreads();                             // done with buffer kc (reused kc+2)
    }

#pragma unroll
    for (int i = 0; i < 4; ++i) {
        const int n0 = nbase + wave * 64 + i * 16;
#pragma unroll
        for (int r = 0; r < 8; ++r) {
            const int row = m0 + r + 8 * half;       // C/D layout: VGPR r -> M=r (+8 hi half)
            xz[(size_t)row * E2c + n0 + lm] = c[i][r];
        }
    }
}

// ---------------- K2: depthwise causal conv (K=4) + bias + SiLU ----------------
__global__ void k2_conv(const float* __restrict__ xz,
                        const float* __restrict__ conv_w,
                        const float* __restrict__ conv_b,
                        float* __restrict__ xc) {
    const int gid = blockIdx.x * 256 + threadIdx.x;
    if (gid >= MTOTc * D_INNERc) return;
    const int d = gid & (D_INNERc - 1);
    const int m = gid >> 9;
    const int t = m & (TTc - 1);
    const int b = m >> 6;
    float acc = conv_b[d];
#pragma unroll
    for (int k = 0; k < D_CONVc; ++k) {
        const int tt = t - (D_CONVc - 1) + k;
        if (tt >= 0)
            acc += xz[(size_t)(b * TTc + tt) * E2c + d] * conv_w[d * D_CONVc + k];
    }
    xc[gid] = silu_f(acc);
}

// ---------------- K3: x_dbl = xc @ W_xproj^T ; dt = softplus(dt_in @ W_dt^T + b_dt) ----------------
__global__ void k3_xproj(const float* __restrict__ xc,
                         const float* __restrict__ W_xproj,
                         const float* __restrict__ W_dt,
                         const float* __restrict__ b_dt,
                         float* __restrict__ xdbl,
                         float* __restrict__ dtb) {
    __shared__ float xcs[D_INNERc];
    __shared__ float dts[DT_RANKc];
    const int m   = blockIdx.x;
    const int tid = threadIdx.x;
#pragma unroll
    for (int i = 0; i < 4; ++i)
        xcs[tid * 4 + i] = xc[(size_t)m * D_INNERc + tid * 4 + i];
    __syncthreads();
    if (tid < XDBLc) {
        const float* wrow = W_xproj + (size_t)tid * D_INNERc;
        float acc = 0.f;
        for (int k = 0; k < D_INNERc; ++k) acc += xcs[k] * wrow[k];
        xdbl[(size_t)m * XDBLc + tid] = acc;
        if (tid < DT_RANKc) dts[tid] = acc;
    }
    __syncthreads();
#pragma unroll
    for (int rep = 0; rep < 4; ++rep) {
        const int d = tid + rep * 128;
        const float* wrow = W_dt + (size_t)d * DT_RANKc;
        float acc = b_dt[d];
#pragma unroll
        for (int r = 0; r < DT_RANKc; ++r) acc += dts[r] * wrow[r];
        dtb[(size_t)m * D_INNERc + d] = (acc > 20.f) ? acc : log1pf(__expf(acc));
    }
}

// ---------------- K4: selective scan + skip + SiLU(z) gating ----------------
__global__ void k4_scan(const float* __restrict__ xc,
                        const float* __restrict__ xdbl,
                        const float* __restrict__ dtb,
                        const float* __restrict__ A_log,
                        const float* __restrict__ D_skip,
                        float* __restrict__ xz /* z in hi half; y out lo half */) {
    __shared__ float bc[2 * D_STATEc];
    const int b = blockIdx.x;
    const int d = threadIdx.x;          // 0..511
    float Arow[D_STATEc], h[D_STATEc];
#pragma unroll
    for (int s = 0; s < D_STATEc; ++s) {
        Arow[s] = -__expf(A_log[d * D_STATEc + s]);
        h[s]    = 0.f;
    }
    const float dsk = D_skip[d];
    for (int t = 0; t < TTc; ++t) {
        const int m = b * TTc + t;
        __syncthreads();                 // protect bc[] from previous iteration
        if (d < 2 * D_STATEc) bc[d] = xdbl[(size_t)m * XDBLc + DT_RANKc + d];
        __syncthreads();
        const float dtv = dtb[(size_t)m * D_INNERc + d];
        const float xcv = xc[(size_t)m * D_INNERc + d];
        const float zv  = xz[(size_t)m * E2c + D_INNERc + d];
        const float du  = dtv * xcv;
        float y = 0.f;
#pragma unroll
        for (int s = 0; s < D_STATEc; ++s) {
            h[s] = __expf(dtv * Arow[s]) * h[s] + du * bc[s];
            y   += h[s] * bc[D_STATEc + s];
        }
        y += xcv * dsk;
        y *= silu_f(zv);
        xz[(size_t)m * E2c + d] = y;     // overwrite dead xc-pre region
    }
}

// ---------------- K5: out = RMSNorm(y @ W_out^T) * ln_w + residual ----------------
__global__ void k5_gemm_out(const float* __restrict__ xz /* y, ld=1024 */,
                            const float* __restrict__ W_out,
                            const float* __restrict__ ln_w,
                            const float* __restrict__ x,
                            float* __restrict__ out) {
    __shared__ float tileS[16][D_MODELc + 4];
    __shared__ float red[16][16];
    __shared__ float rs[16];
    const int lane = threadIdx.x & 31;
    const int wave = threadIdx.x >> 5;
    const int half = lane >> 4, lm = lane & 15;
    const int m0   = blockIdx.x * 16;

    const float* ap  = xz + (size_t)(m0 + lm) * E2c + 2 * half;
    const int n0a = wave * 16, n0b = wave * 16 + 128;
    const float* bpa = W_out + (size_t)(n0a + lm) * D_INNERc + 2 * half;
    const float* bpb = W_out + (size_t)(n0b + lm) * D_INNERc + 2 * half;

    v8f c0 = {0.f,0.f,0.f,0.f,0.f,0.f,0.f,0.f};
    v8f c1 = {0.f,0.f,0.f,0.f,0.f,0.f,0.f,0.f};
#pragma unroll 4
    for (int kk = 0; kk < D_INNERc; kk += 4) {
        v2f a  = *(const v2f*)(ap  + kk);
        v2f b0 = *(const v2f*)(bpa + kk);
        v2f b1 = *(const v2f*)(bpb + kk);
        c0 = wmma_f32x4(a, b0, c0);
        c1 = wmma_f32x4(a, b1, c1);
    }
#pragma unroll
    for (int i = 0; i < 8; ++i) {
        tileS[i + 8 * half][n0a + lm] = c0[i];
        tileS[i + 8 * half][n0b + lm] = c1[i];
    }
    __syncthreads();

    // fused RMSNorm over the 256-wide row + ln_w + residual + (N,T,J,C) store
    const int row = threadIdx.x >> 4;
    const int sub = threadIdx.x & 15;
    float p = 0.f;
#pragma unroll
    for (int i = 0; i < 16; ++i) {
        const float v = tileS[row][sub * 16 + i];
        p += v * v;
    }
    red[row][sub] = p;
    __syncthreads();
    if (sub == 0) {
        float s = 0.f;
#pragma unroll
        for (int i = 0; i < 16; ++i) s += red[row][i];
        rs[row] = rsqrtf(s * (1.0f / D_MODELc) + 1e-6f);
    }
    __syncthreads();
    const int   base  = x_base(m0 + row);
    const float scale = rs[row];
#pragma unroll
    for (int i = 0; i < 16; ++i) {
        const int col = sub * 16 + i;
        out[base + col] = tileS[row][col] * scale * ln_w[col] + x[base + col];
    }
}

// ---------------- launcher ----------------
extern "C" void kernel_launch(void* const* d_in, const int* in_sizes, int n_in,
                              void* d_out, int out_size, void* d_ws, size_t ws_size,
                              hipStream_t stream) {
    const float* x       = (const float*)d_in[0];
    const float* W_in    = (const float*)d_in[1];
    const float* conv_w  = (const float*)d_in[2];
    const float* conv_b  = (const float*)d_in[3];
    const float* W_xproj = (const float*)d_in[4];
    const float* W_dt    = (const float*)d_in[5];
    const float* b_dt    = (const float*)d_in[6];
    const float* A_log   = (const float*)d_in[7];
    const float* D_skip  = (const float*)d_in[8];
    const float* W_out   = (const float*)d_in[9];
    const float* ln_w    = (const float*)d_in[10];
    float* out = (float*)d_out;

    float* ws   = (float*)d_ws;
    float* xz   = ws;                               // MTOT*1024 (xc_pre | z) -> later (y | z)
    float* xc   = xz   + (size_t)MTOTc * E2c;       // MTOT*512
    float* xdbl = xc   + (size_t)MTOTc * D_INNERc;  // MTOT*48
    float* dtb  = xdbl + (size_t)MTOTc * XDBLc;     // MTOT*512

    k1_gemm_in<<<(MTOTc / 16) * 2, 256, 65536, stream>>>(x, W_in, xz);
    k2_conv<<<(MTOTc * D_INNERc) / 256, 256, 0, stream>>>(xz, conv_w, conv_b, xc);
    k3_xproj<<<MTOTc, 128, 0, stream>>>(xc, W_xproj, W_dt, b_dt, xdbl, dtb);
    k4_scan<<<BSEQc, D_INNERc, 0, stream>>>(xc, xdbl, dtb, A_log, D_skip, xz);
    k5_gemm_out<<<MTOTc / 16, 256, 0, stream>>>(xz, W_out, ln_w, x, out);
}